// ClasswiseECE_33303176413864
// MI455X (gfx1250) — compile-verified
//
#include <hip/hip_runtime.h>

// Classwise ECE for N=131072 samples, C=1000 classes, 15 bins.
// Bandwidth-bound: single fused pass over 524MB of logits (~22.5us floor at
// 23.3 TB/s on MI455X). Rows are staged global->LDS with gfx1250 async loads
// (ASYNCcnt) and double-buffered per wave; softmax + binning happen from
// registers. Bin-0 stats (which dominate: max softmax prob << 1/15 here) are
// recovered algebraically, so almost no atomics are needed.

#define ECE_N     131072
#define ECE_C     1000
#define ECE_BINS  15
#define ROWBYTES  4000            // 1000 f32 per row
#define WPB       8               // waves per block (wave32)
#define TPB       (WPB * 32)
#define MAIN_BLOCKS 512

// Issue 8 async b128 loads (32 lanes x 16B = 512B each) covering one row
// (4000B) from global memory into this wave's LDS buffer. Lanes whose chunk-7
// slot would run past the row end are clamped to the row start (their LDS
// slots are never consumed).
__device__ __forceinline__ void async_row_to_lds(const float* logits, int row,
                                                 unsigned lds_base, int lane) {
  unsigned long long g0 = (unsigned long long)(const void*)logits +
                          (unsigned long long)row * (unsigned long long)ROWBYTES;
  unsigned lane16 = (unsigned)lane * 16u;
#pragma unroll
  for (int i = 0; i < 8; ++i) {
    unsigned lo = (unsigned)i * 512u + lane16;
    unsigned go = (lo <= (unsigned)(ROWBYTES - 16)) ? lo : 0u;
    unsigned laddr = lds_base + lo;
    unsigned long long gaddr = g0 + (unsigned long long)go;
    asm volatile("global_load_async_to_lds_b128 %0, %1, off"
                 :: "v"(laddr), "v"(gaddr)
                 : "memory");
  }
}

__global__ __launch_bounds__(TPB) void ece_main_kernel(
    const float* __restrict__ logits, const int* __restrict__ labels,
    float* __restrict__ gcount, float* __restrict__ gconf,
    float* __restrict__ gacc, float* __restrict__ gconfall) {
  // 8 waves * 2 buffers * 4000B = 64000B; reused as the 1000-float block
  // reduction array after the row loop.
  __shared__ __align__(16) float stage[WPB * 2 * ECE_C];

  const int lane = (int)(threadIdx.x & 31u);
  const int wv   = (int)(threadIdx.x >> 5);
  const int gw   = (int)blockIdx.x * WPB + wv;           // global wave id
  const int nw   = (int)gridDim.x * WPB;                 // total waves
  const unsigned lds0 =
      (unsigned)(unsigned long long)(const void*)&stage[wv * 2 * ECE_C];
  const float NEG_INF = -__builtin_inff();

  // Per-lane register accumulators: sum of p over all processed rows for the
  // 32 classes this lane owns (c = 128*k + 4*lane + j).
  float cf[32];
#pragma unroll
  for (int s = 0; s < 32; ++s) cf[s] = 0.f;

  if (gw < ECE_N) async_row_to_lds(logits, gw, lds0, lane);
  int cur = 0;

  for (int r = gw; r < ECE_N; r += nw) {
    int rn = r + nw;
    if (rn < ECE_N) {
      // Prefetch next row into the other buffer, then wait only for the
      // current row's 8 async loads (async loads complete in order).
      async_row_to_lds(logits, rn, lds0 + (unsigned)((cur ^ 1) * ROWBYTES), lane);
      asm volatile("s_wait_asynccnt 8" ::: "memory");
    } else {
      asm volatile("s_wait_asynccnt 0" ::: "memory");
    }

    const float4* lp = (const float4*)&stage[(wv * 2 + cur) * ECE_C];
    float va[32];
#pragma unroll
    for (int i = 0; i < 8; ++i) {
      float4 q = lp[i * 32 + lane];
      va[4 * i + 0] = q.x; va[4 * i + 1] = q.y;
      va[4 * i + 2] = q.z; va[4 * i + 3] = q.w;
    }
    // Chunk 7 covers classes 896..1023; lanes >= 26 hold classes >= 1000.
    if (lane >= 26) { va[28] = NEG_INF; va[29] = NEG_INF;
                      va[30] = NEG_INF; va[31] = NEG_INF; }

    // Row max (lane-local then wave32 shuffle tree).
    float m = NEG_INF;
#pragma unroll
    for (int s = 0; s < 32; ++s) m = fmaxf(m, va[s]);
#pragma unroll
    for (int o = 16; o > 0; o >>= 1) m = fmaxf(m, __shfl_xor(m, o, 32));

    // exp and row sum; keep t = exp(x - m) in registers.
    float ss = 0.f;
#pragma unroll
    for (int s = 0; s < 32; ++s) { float t = __expf(va[s] - m); va[s] = t; ss += t; }
#pragma unroll
    for (int o = 16; o > 0; o >>= 1) ss += __shfl_xor(ss, o, 32);
    float inv = 1.0f / ss;

    int lab = labels[r];

#pragma unroll
    for (int s = 0; s < 32; ++s) {
      float p = va[s] * inv;
      cf[s] += p;                                   // Sigma_all p (register)
      int c = ((s >> 2) << 7) + (lane << 2) + (s & 3);
      float q15 = p * 15.f;
      bool hi  = q15 > 1.f;                         // bin >= 1 (rare)
      bool isl = (c == lab);                        // one per row
      if (hi || isl) {
        int b = (int)ceilf(q15) - 1;
        b = b < 0 ? 0 : (b > 14 ? 14 : b);
        int base = c * ECE_BINS + b;
        if (hi)  { atomicAdd(&gcount[base], 1.f); atomicAdd(&gconf[base], p); }
        if (isl) { atomicAdd(&gacc[base], 1.f); }
      }
    }
    cur ^= 1;
  }

  // Block-level reduction of per-lane conf sums: reuse staging LDS.
  __syncthreads();
  for (int i = (int)threadIdx.x; i < ECE_C; i += TPB) stage[i] = 0.f;
  __syncthreads();
#pragma unroll
  for (int s = 0; s < 32; ++s) {
    int c = ((s >> 2) << 7) + (lane << 2) + (s & 3);
    if (c < ECE_C && cf[s] != 0.f) atomicAdd(&stage[c], cf[s]);
  }
  __syncthreads();
  for (int c = (int)threadIdx.x; c < ECE_C; c += TPB) {
    float v = stage[c];
    if (v != 0.f) atomicAdd(&gconfall[c], v);       // 1000 atomics per block
  }
}

__global__ void ece_final_kernel(const float* __restrict__ gcount,
                                 const float* __restrict__ gconf,
                                 const float* __restrict__ gacc,
                                 const float* __restrict__ gconfall,
                                 float* __restrict__ out) {
  int c = blockIdx.x * blockDim.x + threadIdx.x;
  if (c >= ECE_C) return;
  const float invN = 1.0f / (float)ECE_N;
  float sumCnt = 0.f, sumConf = 0.f, sce = 0.f;
#pragma unroll
  for (int b = 1; b < ECE_BINS; ++b) {
    float cb = gcount[c * ECE_BINS + b];
    float fb = gconf[c * ECE_BINS + b];
    sumCnt += cb; sumConf += fb;
    if (cb > 0.5f) {
      float ab = gacc[c * ECE_BINS + b];
      sce += fabsf((fb - ab) / cb) * (cb * invN);   // |conf/n - acc/n| * n/N
    }
  }
  // Bin 0 recovered algebraically (counts are exact integers in f32).
  float cnt0 = (float)ECE_N - sumCnt;
  if (cnt0 > 0.5f) {
    float conf0 = gconfall[c] - sumConf;
    float acc0  = gacc[c * ECE_BINS + 0];
    sce += fabsf((conf0 - acc0) / cnt0) * (cnt0 * invN);
  }
  atomicAdd(out, sce * (1.0f / (float)ECE_C));
}

__global__ void ece_zero_kernel(float* __restrict__ ws, float* __restrict__ out) {
  int i = blockIdx.x * blockDim.x + threadIdx.x;
  if (i < 46000) ws[i] = 0.f;   // gcount/gconf/gacc (3*15000) + gconfall (1000)
  if (i == 0) out[0] = 0.f;
}

extern "C" void kernel_launch(void* const* d_in, const int* in_sizes, int n_in,
                              void* d_out, int out_size, void* d_ws, size_t ws_size,
                              hipStream_t stream) {
  (void)in_sizes; (void)n_in; (void)out_size; (void)ws_size;
  const float* logits = (const float*)d_in[0];
  const int*   labels = (const int*)d_in[1];
  float* out = (float*)d_out;
  float* ws  = (float*)d_ws;
  float* gcount   = ws;            // [1000][15]
  float* gconf    = ws + 15000;    // [1000][15]
  float* gacc     = ws + 30000;    // [1000][15]
  float* gconfall = ws + 45000;    // [1000]

  ece_zero_kernel<<<(46000 + 255) / 256, 256, 0, stream>>>(ws, out);
  ece_main_kernel<<<MAIN_BLOCKS, TPB, 0, stream>>>(logits, labels, gcount,
                                                   gconf, gacc, gconfall);
  ece_final_kernel<<<(ECE_C + 255) / 256, 256, 0, stream>>>(gcount, gconf,
                                                            gacc, gconfall, out);
}